// KGEmbeddingLayer_58772332478973
// MI455X (gfx1250) — compile-verified
//
#include <hip/hip_runtime.h>

typedef __attribute__((ext_vector_type(2))) float v2f;
typedef __attribute__((ext_vector_type(8))) float v8f;

#define DDIM   128
#define NEIGH  16
#define KNEG   4
#define WAVES  8          // waves per block (256 threads, wave32)
#define MTRIP  8192       // B * NEIGH
#define BSZ    512
#define NCOL   16         // padded B columns (6 live + 10 zero)
#define SLAB   ((NCOL + 1) * DDIM)   // + rel_e column

__global__ void kg_zero_out(float* out) {
    if (threadIdx.x == 0 && blockIdx.x == 0) out[0] = 0.0f;
}

__global__ __launch_bounds__(256)
void kg_loss_wmma(const float* __restrict__ mol_emb,
                  const float* __restrict__ entity_emb,
                  const float* __restrict__ relation_emb,
                  const float* __restrict__ relation_matrix,
                  const int*   __restrict__ tail_ids,
                  const int*   __restrict__ rel_ids,
                  const int*   __restrict__ neg_indices,
                  float*       __restrict__ out)
{
    // per-wave slab: cols 0..5 = RHS vectors (self, tail, neg0..3),
    // cols 6..15 = zeros (B padding), col 16 = rel_e
    __shared__ float lds[WAVES][SLAB];

    const int tid  = threadIdx.x;
    const int w    = tid >> 5;
    const int lane = tid & 31;
    const int m    = blockIdx.x * WAVES + w;

    const int t = tail_ids[m];
    const int r = rel_ids[m];

    float* slab = &lds[w][0];

    // ---- stage RHS vectors + zero padding + rel_e into LDS ----
    {
        const float* src_self = mol_emb      + (size_t)(m >> 4) * DDIM;  // m / NEIGH
        const float* src_tail = entity_emb   + (size_t)t * DDIM;
        const float* src_rele = relation_emb + (size_t)r * DDIM;
        #pragma unroll
        for (int i = lane; i < DDIM; i += 32) {
            slab[0 * DDIM + i]    = src_self[i];
            slab[1 * DDIM + i]    = src_tail[i];
            slab[NCOL * DDIM + i] = src_rele[i];
        }
        #pragma unroll
        for (int k = 0; k < KNEG; ++k) {
            const int nm = neg_indices[(size_t)m * KNEG + k];
            const int nt = tail_ids[nm];
            const float* s = entity_emb + (size_t)nt * DDIM;
            #pragma unroll
            for (int i = lane; i < DDIM; i += 32)
                slab[(2 + k) * DDIM + i] = s[i];
        }
        // zero-fill padding columns 6..15 once (kills per-step EXEC masking)
        #pragma unroll
        for (int i = lane; i < 10 * DDIM; i += 32)
            slab[6 * DDIM + i] = 0.0f;
    }
    __syncthreads();

    const float* Rm  = relation_matrix + (size_t)r * DDIM * DDIM;  // L2-resident
    const int half = lane >> 4;   // 0: K={0,1}, 1: K={2,3} within each K=4 step
    const int sub  = lane & 15;   // row (A) / column (B,C)

    // 8 row-tile accumulators = full 128x16 output
    v8f c[8] = {};

    const float* bbase = slab + sub * DDIM + 2 * half;             // always valid (padded)
    const float* abase = Rm + (size_t)sub * DDIM + 2 * half;

    for (int kk = 0; kk < DDIM / 4; ++kk) {
        const v2f b = *(const v2f*)(bbase + 4 * kk);               // ds_load_b64, unconditional
        #pragma unroll
        for (int mt = 0; mt < 8; ++mt) {
            const v2f a = *(const v2f*)(abase + (size_t)(16 * mt) * DDIM + 4 * kk); // global_load_b64
            c[mt] = __builtin_amdgcn_wmma_f32_16x16x4_f32(
                        /*neg_a=*/false, a, /*neg_b=*/false, b,
                        /*c_mod=*/(short)0, c[mt],
                        /*reuse_a=*/false, /*reuse_b=*/false);
        }
    }

    // ---- score phase ----
    // C layout: VGPR j -> row 16*mt + j + 8*half, column = sub.
    // col0 = Rh, col1 = Rt, col2..5 = Rt_neg. u = Rh + rel_e; acc_n = sum (u - col_n)^2
    float acc = 0.0f;
    const int srcl = half << 4;   // lane holding column 0 of our rows (0 or 16)
    #pragma unroll
    for (int mt = 0; mt < 8; ++mt) {
        #pragma unroll
        for (int j = 0; j < 8; ++j) {
            const int row   = 16 * mt + j + 8 * half;
            const float rel = slab[NCOL * DDIM + row];
            const float rh  = __shfl(c[mt][j], srcl, 32);
            const float d   = (rh + rel) - c[mt][j];
            acc += d * d;
        }
    }

    // combine column halves: score_n = acc@lane n + acc@lane n+16
    const float s1 = __shfl(acc, 1, 32) + __shfl(acc, 17, 32);  // pos (Rt)
    const float s2 = __shfl(acc, 2, 32) + __shfl(acc, 18, 32);
    const float s3 = __shfl(acc, 3, 32) + __shfl(acc, 19, 32);
    const float s4 = __shfl(acc, 4, 32) + __shfl(acc, 20, 32);
    const float s5 = __shfl(acc, 5, 32) + __shfl(acc, 21, 32);

    const float pos  = s1;
    const float negm = 0.25f * (s2 + s3 + s4 + s5);
    const float z    = negm - pos;
    const float sig  = 1.0f / (1.0f + __expf(-z));
    const float loss = -__logf(sig + 1e-12f);

    if (lane == 0)
        atomicAdd(out, loss * (1.0f / (float)BSZ));
}

extern "C" void kernel_launch(void* const* d_in, const int* in_sizes, int n_in,
                              void* d_out, int out_size, void* d_ws, size_t ws_size,
                              hipStream_t stream) {
    (void)in_sizes; (void)n_in; (void)out_size; (void)d_ws; (void)ws_size;
    const float* mol  = (const float*)d_in[0];
    const float* ent  = (const float*)d_in[1];
    const float* rele = (const float*)d_in[2];
    const float* relm = (const float*)d_in[3];
    const int*   tids = (const int*)d_in[4];
    const int*   rids = (const int*)d_in[5];
    const int*   negi = (const int*)d_in[6];
    float* out = (float*)d_out;

    kg_zero_out<<<dim3(1), dim3(1), 0, stream>>>(out);
    kg_loss_wmma<<<dim3(MTRIP / WAVES), dim3(WAVES * 32), 0, stream>>>(
        mol, ent, rele, relm, tids, rids, negi, out);
}